// GIN_240518168949
// MI455X (gfx1250) — compile-verified
//
#include <hip/hip_runtime.h>

typedef float v2f __attribute__((ext_vector_type(2)));
typedef float v8f __attribute__((ext_vector_type(8)));
typedef unsigned v4u __attribute__((ext_vector_type(4)));
typedef unsigned v8u __attribute__((ext_vector_type(8)));

#define HD 128        // hidden / input feature dim
#define KP 132        // padded LDS stride for A tile (128 + 4 dwords TDM pad)
#define GIN_BN_EPS 1e-5f

// ---------------------------------------------------------------------------
// agg = x  (initialize aggregation buffer with self term, float4 vectorized)
// ---------------------------------------------------------------------------
__global__ void gin_copy_f4(const float* __restrict__ src, float* __restrict__ dst, int n4) {
  int i = blockIdx.x * blockDim.x + threadIdx.x;
  if (i < n4) ((float4*)dst)[i] = ((const float4*)src)[i];
}

// ---------------------------------------------------------------------------
// agg[dst] += x[src]   — one wave per edge: lanes cover 32 x float4 = 128 ch.
// src/dst index loads are wave-uniform -> scalar loads; gather is one
// coalesced 512B row; scatter via global_atomic_add_f32 (L2-resident rows).
// ---------------------------------------------------------------------------
__global__ void gin_scatter(const float* __restrict__ x,
                            const long long* __restrict__ src,
                            const long long* __restrict__ dst,
                            float* __restrict__ agg, int E) {
  int t = blockIdx.x * blockDim.x + threadIdx.x;
  int e = t >> 5;
  if (e >= E) return;
  int c = (t & 31) << 2;                       // channel group (4 floats)
  int s = (int)src[e];
  int d = (int)dst[e];
  const float4 v = *(const float4*)(x + (size_t)s * HD + c);
  float* p = agg + (size_t)d * HD + c;
  atomicAdd(p + 0, v.x);
  atomicAdd(p + 1, v.y);
  atomicAdd(p + 2, v.z);
  atomicAdd(p + 3, v.w);
}

// ---------------------------------------------------------------------------
// TDM: issue one tensor_load_to_lds for a 2D tile of 4-byte elements.
//   gptr      : global address of tile start (row-major, row stride = dim0)
//   lds_off   : LDS byte address of destination
//   dim0/dim1 : tensor extents (elements); rows >= dim1 read as zero (OOB)
//   tile0/1   : tile extents (elements)
//   pi/pa     : D# pad_interval code (dwords: 0=2..7=256), pad_amount code
//               (dwords: amount-1)  -> LDS row stride = tile0 + (pa+1) dwords
// Descriptor layout per CDNA5 ISA.
// ---------------------------------------------------------------------------
__device__ __forceinline__ void tdm_load_2d(const void* gptr, unsigned lds_off,
                                            unsigned dim0, unsigned dim1,
                                            unsigned tile0, unsigned tile1,
                                            unsigned pi, unsigned pa) {
  unsigned long long ga = (unsigned long long)(uintptr_t)gptr;
  v4u g0;
  g0.x = 1u;                                            // count=1, user mode
  g0.y = lds_off;                                       // lds_addr
  g0.z = (unsigned)ga;                                  // global_addr[31:0]
  g0.w = (unsigned)((ga >> 32) & 0x1FFFFFFu)            // global_addr[56:32]
         | 0x80000000u;                                 // type=2 ("image")
  v8u g1;
  g1[0] = (2u << 16)                                    // data_size = 4 bytes
          | (1u << 20)                                  // pad_enable
          | (pi << 22) | (pa << 25);                    // pad interval/amount
  g1[1] = (dim0 & 0xFFFFu) << 16;                       // tensor_dim0 lo16
  g1[2] = (dim0 >> 16) | ((dim1 & 0xFFFFu) << 16);      // dim0 hi / dim1 lo
  g1[3] = (dim1 >> 16) | (tile0 << 16);                 // dim1 hi / tile_dim0
  g1[4] = tile1;                                        // tile_dim1 (dim2=0)
  g1[5] = dim0;                                         // tensor_dim0_stride lo
  g1[6] = 0u;                                           // stride0 hi / stride1 lo
  g1[7] = 0u;
  asm volatile("tensor_load_to_lds %0, %1" :: "s"(g0), "s"(g1) : "memory");
}

// ---------------------------------------------------------------------------
// out[M x N] = epilogue(A[M x 128] @ W[128 x N])
//   BN_RELU: y = relu((acc + bias) * s + t)  with s,t folded BN params
//   else:    y = acc + bias
// 8 waves/block; each wave computes one 16x16 tile with v_wmma_f32_16x16x4_f32.
// W panel + A row-tile staged in LDS by the Tensor Data Mover (wave 0 issues,
// TENSORcnt wait, workgroup barrier publishes). TDM hardware padding gives
// bank-conflict-free strides; TDM OOB-zero handles the M tail.
// ---------------------------------------------------------------------------
template <int TILES_N, bool BN_RELU>
__global__ void gin_gemm(const float* __restrict__ A, const float* __restrict__ W,
                         const float* __restrict__ bias,
                         const float* __restrict__ gam, const float* __restrict__ bet,
                         const float* __restrict__ mu, const float* __restrict__ var,
                         float* __restrict__ out, int M) {
  constexpr int N = TILES_N * 16;
  constexpr int NP = N + 8;                 // 2*NP % 64 == 16 -> disjoint bank groups
  constexpr int ROW_TILES = 8 / TILES_N;    // row tiles per block
  constexpr int BM = ROW_TILES * 16;        // rows per block
  constexpr unsigned PI_W = (N == 128) ? 6u : 5u;  // pad every N dwords
  constexpr unsigned PA_W = 7u;                    // +8 dwords  -> stride NP
  constexpr unsigned PI_A = 6u;                    // pad every 128 dwords
  constexpr unsigned PA_A = 3u;                    // +4 dwords  -> stride KP

  extern __shared__ float smem[];
  float* sW = smem;                         // HD x NP
  float* sA = smem + HD * NP;               // BM x KP

  const int tid = threadIdx.x;
  const int lane = tid & 31;
  const int wave = tid >> 5;
  const int row0 = blockIdx.x * BM;

  // ---- stage W (HD x N) and A tile (BM x HD) via Tensor Data Mover
  if (wave == 0) {
    tdm_load_2d(W, (unsigned)(uintptr_t)sW, N, HD, N, HD, PI_W, PA_W);
    tdm_load_2d(A + (size_t)row0 * HD, (unsigned)(uintptr_t)sA,
                HD, (unsigned)(M - row0), HD, BM, PI_A, PA_A);
    __builtin_amdgcn_s_wait_tensorcnt(0);
  }
  __syncthreads();

  const int tile_n = wave % TILES_N;
  const int tile_m = wave / TILES_N;
  const int nloc = lane & 15;               // column within 16-wide tile
  const int ncol = tile_n * 16 + nloc;      // global column
  const int arow = tile_m * 16 + nloc;      // A row for this lane (A frag: lane = M)
  const int kk = (lane >> 4) * 2;           // half-wave K split: {k0,k0+1} vs {k0+2,k0+3}

  v8f acc = {};
#pragma unroll
  for (int k0 = 0; k0 < HD; k0 += 4) {
    v2f a, b;
    a.x = sA[arow * KP + k0 + kk];
    a.y = sA[arow * KP + k0 + kk + 1];
    b.x = sW[(k0 + kk) * NP + ncol];
    b.y = sW[(k0 + kk + 1) * NP + ncol];
    acc = __builtin_amdgcn_wmma_f32_16x16x4_f32(
        /*neg_a=*/false, a, /*neg_b=*/false, b,
        /*c_mod=*/(short)0, acc, /*reuse_a=*/false, /*reuse_b=*/false);
  }

  // ---- fused epilogue: fold bias + BN into per-column scale/shift
  float scale = 1.f, shift = bias[ncol];
  if (BN_RELU) {
    float s = gam[ncol] * rsqrtf(var[ncol] + GIN_BN_EPS);
    shift = bet[ncol] - mu[ncol] * s + bias[ncol] * s;
    scale = s;
  }

  // C/D layout: VGPR r, lanes 0-15 -> M=r, lanes 16-31 -> M=r+8; N = lane&15
  const int mbase = tile_m * 16 + (lane >> 4) * 8;
#pragma unroll
  for (int r = 0; r < 8; ++r) {
    int gm = row0 + mbase + r;
    if (gm < M) {
      float y = fmaf(acc[r], scale, shift);
      if (BN_RELU) y = fmaxf(y, 0.f);
      out[(size_t)gm * N + ncol] = y;
    }
  }
}

// ---------------------------------------------------------------------------
extern "C" void kernel_launch(void* const* d_in, const int* in_sizes, int n_in,
                              void* d_out, int out_size, void* d_ws, size_t ws_size,
                              hipStream_t stream) {
  const float* x = (const float*)d_in[0];
  const long long* ei = (const long long*)d_in[1];
  const float* W0 = (const float*)d_in[2];
  const float* b0 = (const float*)d_in[3];
  const float* g0 = (const float*)d_in[4];
  const float* be0 = (const float*)d_in[5];
  const float* m0 = (const float*)d_in[6];
  const float* v0 = (const float*)d_in[7];
  const float* W1 = (const float*)d_in[8];
  const float* b1 = (const float*)d_in[9];
  const float* g1 = (const float*)d_in[10];
  const float* be1 = (const float*)d_in[11];
  const float* m1 = (const float*)d_in[12];
  const float* v1 = (const float*)d_in[13];
  const float* W2 = (const float*)d_in[14];
  const float* b2 = (const float*)d_in[15];
  const float* g2 = (const float*)d_in[16];
  const float* be2 = (const float*)d_in[17];
  const float* m2 = (const float*)d_in[18];
  const float* v2 = (const float*)d_in[19];
  const float* W3 = (const float*)d_in[20];
  const float* b3 = (const float*)d_in[21];

  const int Nn = in_sizes[0] / HD;          // 50000 nodes
  const int E = in_sizes[1] / 2;            // 800000 edges
  const long long* src = ei;
  const long long* dst = ei + E;

  float* buf0 = (float*)d_ws;                       // agg buffer
  float* buf1 = buf0 + (size_t)Nn * HD;             // hidden buffer

  const dim3 blk(256);
  const int n4 = Nn * HD / 4;
  const dim3 gCopy((n4 + 255) / 256);
  const dim3 gScat(((size_t)E * 32 + 255) / 256);
  const dim3 gG128((Nn + 15) / 16);
  const dim3 gG64((Nn + 31) / 32);
  // +64 dwords slack: TDM pads after the final row too
  const size_t smem128 = (size_t)(HD * (128 + 8) + 16 * KP + 64) * sizeof(float);
  const size_t smem64 = (size_t)(HD * (64 + 8) + 32 * KP + 64) * sizeof(float);

  // ---- layer 0: agg(x) -> relu(bn(agg @ W0 + b0))   (relu∘relu == relu)
  gin_copy_f4<<<gCopy, blk, 0, stream>>>(x, buf0, n4);
  gin_scatter<<<gScat, blk, 0, stream>>>(x, src, dst, buf0, E);
  gin_gemm<8, true><<<gG128, blk, smem128, stream>>>(buf0, W0, b0, g0, be0, m0, v0, buf1, Nn);

  // ---- layer 1
  gin_copy_f4<<<gCopy, blk, 0, stream>>>(buf1, buf0, n4);
  gin_scatter<<<gScat, blk, 0, stream>>>(buf1, src, dst, buf0, E);
  gin_gemm<8, true><<<gG128, blk, smem128, stream>>>(buf0, W1, b1, g1, be1, m1, v1, buf1, Nn);

  // ---- layer 2
  gin_copy_f4<<<gCopy, blk, 0, stream>>>(buf1, buf0, n4);
  gin_scatter<<<gScat, blk, 0, stream>>>(buf1, src, dst, buf0, E);
  gin_gemm<8, true><<<gG128, blk, smem128, stream>>>(buf0, W2, b2, g2, be2, m2, v2, buf1, Nn);

  // ---- output projection: out = h2 @ W3 + b3  (raw, no BN/ReLU)
  gin_gemm<4, false><<<gG64, blk, smem64, stream>>>(buf1, W3, b3, nullptr, nullptr, nullptr,
                                                    nullptr, (float*)d_out, Nn);
}